// Net_37031208026813
// MI455X (gfx1250) — compile-verified
//
#include <hip/hip_runtime.h>

typedef __attribute__((ext_vector_type(2))) float v2f;
typedef __attribute__((ext_vector_type(8))) float v8f;

#define B_    128
#define T_    512
#define N_    512
#define IN_   6
#define BT    16      // batch tile = WMMA M
#define HPAD  516     // LDS row pitch (floats): stride 4 banks -> conflict-free
#define KC    128     // 512 / 4 K-chunks per step

// ---------------------------------------------------------------------------
// Pre-pass: pack W_rec (row-major [N][N]) into the exact per-lane VGPR layout
// of the B-operand of V_WMMA_F32_16X16X4_F32.
//   B[k][n] = W_rec[n][k]  (we compute h @ W_rec^T)
//   lane l of tile (ntile, kchunk) holds pair W_rec[ntile*16+(l&15)][4*kc+2*(l>>4) + {0,1}]
// Pack index = ((ntile*128 + kc)*32 + lane)*2  -> hot-loop loads are coalesced b64.
// ---------------------------------------------------------------------------
__global__ void pack_w_kernel(const float* __restrict__ Wrec,
                              float* __restrict__ Wpack) {
    int p    = blockIdx.x * blockDim.x + threadIdx.x;   // 0 .. 131071
    int lane = p & 31;
    int kc   = (p >> 5) & (KC - 1);
    int nt   = p >> 12;                                 // 0 .. 31
    int n    = nt * 16 + (lane & 15);
    int ka   = kc * 4 + 2 * (lane >> 4);
    v2f v;
    v.x = Wrec[n * N_ + ka];
    v.y = Wrec[n * N_ + ka + 1];
    *(v2f*)(Wpack + 2 * (size_t)p) = v;
}

// ---------------------------------------------------------------------------
// Persistent scan kernel: one workgroup per 16-batch tile, h kept in LDS for
// all 511 steps. 512 threads = 16 waves; wave w owns N columns [32w, 32w+32)
// as two 16-wide WMMA tiles. f32 WMMA (16x16x4), K looped 128x per step.
// ---------------------------------------------------------------------------
__global__ void __launch_bounds__(512, 1)
rnn_scan_kernel(const float* __restrict__ u,
                const float* __restrict__ unoise,
                const float* __restrict__ rnoise,
                const float* __restrict__ brec,
                const float* __restrict__ Win,
                const float* __restrict__ Wpack,
                float* __restrict__ out) {
    __shared__ float h_lds[BT * HPAD];   // current state, [16][516]
    __shared__ float uin[BT * 8];        // staged (u + ns*noise) for this t

    const int tid   = threadIdx.x;
    const int wave  = tid >> 5;
    const int lane  = tid & 31;
    const int hi    = lane >> 4;         // 0 | 1 (half-wave)
    const int nl    = lane & 15;
    const int bbase = blockIdx.x * BT;

    const int n0 = wave * 32 + nl;       // tile 0 column
    const int n1 = n0 + 16;              // tile 1 column

    // Per-lane constants: W_in rows + bias for my two columns.
    float w0[IN_], w1[IN_];
#pragma unroll
    for (int i = 0; i < IN_; ++i) {
        w0[i] = Win[n0 * IN_ + i];
        w1[i] = Win[n1 * IN_ + i];
    }
    const float br0 = brec[n0], br1 = brec[n1];

    // Zero-init h and states[:, 0, :].
#pragma unroll
    for (int g = 0; g < 8; ++g) {
        int m = g + 8 * hi;
        h_lds[m * HPAD + n0] = 0.f;
        h_lds[m * HPAD + n1] = 0.f;
        size_t ob = (size_t)(bbase + m) * T_ * N_;
        out[ob + n0] = 0.f;
        out[ob + n1] = 0.f;
    }
    __syncthreads();

    const float noise_scale = 0.632455532033676f;  // sqrt((2/0.2)*0.2^2)
    const float* wp0 = Wpack + (size_t)(wave * 2) * KC * 64;   // tile0 pack base
    const float* wp1 = wp0 + (size_t)KC * 64;                  // tile1 pack base
    const float* arow = h_lds + nl * HPAD + 2 * hi;            // A operand base

    for (int t = 0; t < T_ - 1; ++t) {
        // Stage combined input for this step (96 values: 16 batches x 6).
        if (tid < BT * IN_) {
            int m = tid / IN_, i = tid - m * IN_;
            size_t idx = ((size_t)(bbase + m) * T_ + t) * IN_ + i;
            uin[m * 8 + i] = u[idx] + noise_scale * unoise[idx];
        }

        // h_tile @ W_rec^T via f32 WMMA, K = 512 in chunks of 4.
        v8f c0 = {}, c1 = {};
#pragma unroll 4
        for (int kc = 0; kc < KC; ++kc) {
            v2f a  = *(const v2f*)(arow + kc * 4);                  // LDS, b64
            v2f b0 = *(const v2f*)(wp0 + (kc * 32 + lane) * 2);     // coalesced b64
            v2f b1 = *(const v2f*)(wp1 + (kc * 32 + lane) * 2);
            c0 = __builtin_amdgcn_wmma_f32_16x16x4_f32(
                     false, a, false, b0, (short)0, c0, false, false);
            c1 = __builtin_amdgcn_wmma_f32_16x16x4_f32(
                     false, a, false, b1, (short)0, c1, false, false);
        }
        __syncthreads();   // all reads of h done; uin staged

        // Fused epilogue: input proj + bias + relu + leaky update + noise.
#pragma unroll
        for (int g = 0; g < 8; ++g) {
            int m = g + 8 * hi;           // C layout: lanes>=16 hold M = g+8
            int b = bbase + m;
            float inp0 = 0.f, inp1 = 0.f;
#pragma unroll
            for (int i = 0; i < IN_; ++i) {
                float uv = uin[m * 8 + i];
                inp0 = fmaf(uv, w0[i], inp0);
                inp1 = fmaf(uv, w1[i], inp1);
            }
            size_t rb = ((size_t)b * T_ + t) * N_;
            float rn0 = rnoise[rb + n0];
            float rn1 = rnoise[rb + n1];
            float h0  = h_lds[m * HPAD + n0];
            float h1  = h_lds[m * HPAD + n1];
            float pre0 = c0[g] + br0 + inp0;
            float pre1 = c1[g] + br1 + inp1;
            float hn0 = 0.8f * h0 + 0.2f * (fmaxf(pre0, 0.f) + rn0);
            float hn1 = 0.8f * h1 + 0.2f * (fmaxf(pre1, 0.f) + rn1);
            h_lds[m * HPAD + n0] = hn0;   // each element owned by one thread
            h_lds[m * HPAD + n1] = hn1;
            size_t ob = ((size_t)b * T_ + (t + 1)) * N_;
            out[ob + n0] = hn0;
            out[ob + n1] = hn1;
        }
        __syncthreads();   // h writes visible; uin consumed
    }
}

extern "C" void kernel_launch(void* const* d_in, const int* in_sizes, int n_in,
                              void* d_out, int out_size, void* d_ws, size_t ws_size,
                              hipStream_t stream) {
    const float* u        = (const float*)d_in[0];
    const float* in_noise = (const float*)d_in[1];
    const float* rn       = (const float*)d_in[2];
    const float* Wrec     = (const float*)d_in[3];
    const float* brec     = (const float*)d_in[4];
    const float* Win      = (const float*)d_in[5];
    float* out   = (float*)d_out;
    float* Wpack = (float*)d_ws;     // 1 MB packed W_rec

    pack_w_kernel<<<512, 256, 0, stream>>>(Wrec, Wpack);
    rnn_scan_kernel<<<B_ / BT, 512, 0, stream>>>(u, in_noise, rn, brec, Win,
                                                 Wpack, out);
}